// MultiUTAETemporalExtractor_32358283608073
// MI455X (gfx1250) — compile-verified
//
#include <hip/hip_runtime.h>

typedef float v2f __attribute__((ext_vector_type(2)));
typedef float v4f __attribute__((ext_vector_type(4)));
typedef float v8f __attribute__((ext_vector_type(8)));
typedef int   v4i __attribute__((ext_vector_type(4)));

#define NUM_CLASSES 6
#define B_DIM 16
#define T_DIM 60
#define C_DIM 128
#define NCH   (B_DIM * T_DIM * C_DIM)   /* 122880 channels, 1024 fp32 each */
#define NBT   (B_DIM * T_DIM)           /* 960 (b,t) histogram tiles       */

// ---------------------------------------------------------------------------
// Kernel 1: spatial mean via V_WMMA_F32_16X16X4_F32.
// One wave32 per channel (1024 contiguous floats). A-matrix = all ones, so
// D[m,n] = sum_k B[k,n]: any element->B-slot mapping is valid, letting each
// lane do a coalesced float4 NT load (512B per wave per load) and feed the
// XDL pipe two B fragments per load. D VGPR0 holds per-column sums on all 32
// lanes (duplicated across the two half-wave rows), so the all-lane shfl sum
// equals 2x the channel total -> scale by 0.5/1024.
// ---------------------------------------------------------------------------
__global__ __launch_bounds__(256) void spatial_mean_wmma(
    const float* __restrict__ x, float* __restrict__ emb) {
  const int wave = (int)((blockIdx.x * blockDim.x + threadIdx.x) >> 5);
  const int lane = (int)(threadIdx.x & 31u);
  if (wave >= NCH) return;  // uniform per wave: EXEC stays all-ones for WMMA

  const float* base = x + (size_t)wave * 1024 + (size_t)lane * 4;

  v8f acc = {};
  const v2f ones = {1.0f, 1.0f};

#pragma unroll
  for (int it = 0; it < 8; ++it) {
    // streaming read-once data: non-temporal so we don't churn L2
    v4f d = __builtin_nontemporal_load((const v4f*)(base + it * 128));
    v2f b0 = {d.x, d.y};
    v2f b1 = {d.z, d.w};
    acc = __builtin_amdgcn_wmma_f32_16x16x4_f32(
        /*neg_a=*/false, ones, /*neg_b=*/false, b0,
        /*c_mod=*/(short)0, acc, /*reuse_a=*/false, /*reuse_b=*/false);
    acc = __builtin_amdgcn_wmma_f32_16x16x4_f32(
        false, ones, false, b1, (short)0, acc, false, false);
  }

  float v = acc[0];
#pragma unroll
  for (int off = 16; off > 0; off >>= 1) v += __shfl_xor(v, off, 32);

  if (lane == 0) emb[wave] = v * (0.5f / 1024.0f);
}

// ---------------------------------------------------------------------------
// Kernel 2: per-(b,t) 6-bin histogram + argmax over 256x256 int labels.
// One 256-thread block per (b,t). Each thread streams 64 NT int4 loads
// (contiguous 4KB per block-iteration) and packs per-class counts into a
// 64-bit accumulator (10 bits/class; <=256 pixels/thread so no overflow).
// Warp shfl-reduction, then 6 LDS atomics per wave, then thread 0 argmax
// (ties resolve to lowest class, matching jnp.argmax).
// ---------------------------------------------------------------------------
__global__ __launch_bounds__(256) void patch_majority(
    const int* __restrict__ gt, float* __restrict__ labels) {
  __shared__ int cnt[NUM_CLASSES];
  const int tid = (int)threadIdx.x;
  const int bt  = (int)blockIdx.x;

  if (tid < NUM_CLASSES) cnt[tid] = 0;
  __syncthreads();

  const int* base = gt + (size_t)bt * 65536;

  unsigned long long acc = 0ull;
#pragma unroll 4
  for (int it = 0; it < 64; ++it) {
    v4i p = __builtin_nontemporal_load((const v4i*)(base + it * 1024 + tid * 4));
    acc += 1ull << (p.x * 10);
    acc += 1ull << (p.y * 10);
    acc += 1ull << (p.z * 10);
    acc += 1ull << (p.w * 10);
  }

#pragma unroll
  for (int c = 0; c < NUM_CLASSES; ++c) {
    int v = (int)((acc >> (c * 10)) & 1023ull);
#pragma unroll
    for (int off = 16; off > 0; off >>= 1) v += __shfl_xor(v, off, 32);
    if ((tid & 31) == 0) atomicAdd(&cnt[c], v);
  }
  __syncthreads();

  if (tid == 0) {
    int best = 0;
#pragma unroll
    for (int c = 1; c < NUM_CLASSES; ++c)
      if (cnt[c] > cnt[best]) best = c;
    labels[bt] = (float)best;
  }
}

// ---------------------------------------------------------------------------
// d_in[0]: out_temporal fp32 [16,60,128,32,32]
// d_in[1]: gt int32 [16,60,256,256]
// d_out  : float[122880 emb | 960 labels]
// ---------------------------------------------------------------------------
extern "C" void kernel_launch(void* const* d_in, const int* in_sizes, int n_in,
                              void* d_out, int out_size, void* d_ws, size_t ws_size,
                              hipStream_t stream) {
  (void)in_sizes; (void)n_in; (void)out_size; (void)d_ws; (void)ws_size;
  const float* x  = (const float*)d_in[0];
  const int*   gt = (const int*)d_in[1];
  float* out = (float*)d_out;

  // 122880 channels / 8 waves per 256-thread block = 15360 blocks (exact)
  spatial_mean_wmma<<<NCH / 8, 256, 0, stream>>>(x, out);
  patch_majority<<<NBT, 256, 0, stream>>>(gt, out + NCH);
}